// SimilarityMeasureModel_55851754717413
// MI455X (gfx1250) — compile-verified
//
#include <hip/hip_runtime.h>
#include <math.h>

// Problem constants (from reference): B=32, P=3, D=300, N=2048
#define BB      32
#define PP      3
#define DD      300
#define NN      2048
#define SLABS   (BB * PP)        // 96 independent (b,p) slabs of [D][N]
#define VEC     4                // floats per thread along n (B128 per lane)
#define THREADS 128              // 4 wave32 per block
#define NT      (THREADS * VEC)  // 512 n-columns per block
#define TILES   (NN / NT)        // 4 tiles per slab
#define DCHUNK  10               // d-rows per pipeline stage
#define NSTAGES (DD / DCHUNK)    // 30 stages

static_assert(NN % NT == 0, "n tiling");
static_assert(DD % DCHUNK == 0, "d tiling");
static_assert(4 * DCHUNK <= 63, "ASYNCcnt capacity (2 buffers x 2*DCHUNK)");

// LDS layout: [buf(2)][array(2: x,y)][row(DCHUNK)][col(NT)] floats = 80 KB.
// Single __shared__ object -> allocated at LDS offset 0, so the byte offsets
// we feed to global_load_async_to_lds (VDST vgpr) are computed arithmetically
// (verified in round-1/2 disasm: row stride 0x800, array stride DCHUNK*0x800).
#define SMEM_FLOATS (2 * 2 * DCHUNK * NT)

__global__ __launch_bounds__(THREADS) void sim_kernel(
    const float* __restrict__ x, const float* __restrict__ y,
    float* __restrict__ out) {
  __shared__ float smem[SMEM_FLOATS];

  // CRITICAL: capture smem's address inside inline asm. Without this, LLVM
  // proves the LDS global is never written (the async copies are opaque and
  // only receive integer offsets) and folds every smem load to undef,
  // deleting the whole accumulation loop (observed in round 1: ds=0).
  asm volatile("" :: "s"(&smem[0]) : "memory");

  const int tid  = threadIdx.x;
  const int slab = blockIdx.x / TILES;   // b*P + p
  const int tile = blockIdx.x % TILES;
  const int n0   = tile * NT;

  const float* xs = x + (size_t)slab * DD * NN;  // uniform -> SGPR pair
  const float* ys = y + (size_t)slab * DD * NN;

  // Per-thread byte offsets
  const uint32_t colBytes  = (uint32_t)(n0 + tid * VEC) * 4u;  // global col
  const uint32_t laneBytes = (uint32_t)tid * (VEC * 4u);       // lds col

  // Issue one stage of async global->LDS copies into buffer nb.
  // One instruction moves 32 lanes x 16B = 512B; 2*DCHUNK instructions/stage.
  auto issue = [&](int stage, int nb) {
    const int d0 = stage * DCHUNK;
#pragma unroll
    for (int r = 0; r < DCHUNK; ++r) {
      const uint32_t go = (uint32_t)(d0 + r) * (uint32_t)(NN * 4u) + colBytes;
      const uint32_t lx =
          (uint32_t)(((nb * 2 + 0) * DCHUNK + r) * NT) * 4u + laneBytes;
      const uint32_t ly =
          (uint32_t)(((nb * 2 + 1) * DCHUNK + r) * NT) * 4u + laneBytes;
      // VDST = LDS byte address, VADDR = 32-bit global offset, SADDR = base.
      // Stream is read-once (472MB total, no reuse) -> non-temporal hint.
      asm volatile(
          "global_load_async_to_lds_b128 %0, %1, %2 offset:0 th:TH_LOAD_NT"
          :: "v"(lx), "v"(go), "s"(xs) : "memory");
      asm volatile(
          "global_load_async_to_lds_b128 %0, %1, %2 offset:0 th:TH_LOAD_NT"
          :: "v"(ly), "v"(go), "s"(ys) : "memory");
    }
  };

  float4 dot = make_float4(0.f, 0.f, 0.f, 0.f);
  float4 xx  = dot, yy = dot, l2 = dot, l1 = dot;

  issue(0, 0);  // prologue: fill buffer 0

  for (int s = 0; s < NSTAGES; ++s) {
    const int pb = s & 1;
    if (s + 1 < NSTAGES) {
      // Buffer pb^1 was consumed at stage s-1: make sure those ds reads
      // retired before the DMA overwrites it.
      asm volatile("s_wait_dscnt 0x0" ::: "memory");
      issue(s + 1, pb ^ 1);
      // Allow the 2*DCHUNK just-issued copies to stay in flight; require the
      // previous stage's copies (in-order completion) to be done.
      asm volatile("s_wait_asynccnt %0" :: "i"(2 * DCHUNK) : "memory");
    } else {
      asm volatile("s_wait_asynccnt 0x0" ::: "memory");
    }

    const float4* xb =
        (const float4*)&smem[((pb * 2 + 0) * DCHUNK) * NT] + tid;
    const float4* yb =
        (const float4*)&smem[((pb * 2 + 1) * DCHUNK) * NT] + tid;
#pragma unroll
    for (int r = 0; r < DCHUNK; ++r) {
      const float4 a = xb[r * (NT / VEC)];  // ds_load_b128, conflict-free
      const float4 b = yb[r * (NT / VEC)];
#define ACC(c)                                   \
  {                                              \
    const float av = a.c, bv = b.c;              \
    dot.c = fmaf(av, bv, dot.c);                 \
    xx.c  = fmaf(av, av, xx.c);                  \
    yy.c  = fmaf(bv, bv, yy.c);                  \
    const float df = av - bv;                    \
    l2.c = fmaf(df, df, l2.c);                   \
    l1.c += fabsf(df);                           \
  }
      ACC(x) ACC(y) ACC(z) ACC(w)
#undef ACC
    }
  }

  // Epilogue: cos = dot/(|x||y|), l2 = sqrt(sum d^2), l1 = sum|d|
  float4 cosv, l2v;
  cosv.x = dot.x / (sqrtf(xx.x) * sqrtf(yy.x));
  cosv.y = dot.y / (sqrtf(xx.y) * sqrtf(yy.y));
  cosv.z = dot.z / (sqrtf(xx.z) * sqrtf(yy.z));
  cosv.w = dot.w / (sqrtf(xx.w) * sqrtf(yy.w));
  l2v.x = sqrtf(l2.x); l2v.y = sqrtf(l2.y);
  l2v.z = sqrtf(l2.z); l2v.w = sqrtf(l2.w);

  // out[(slab*N + n)*3 + c]; 4 consecutive n -> 12 contiguous floats,
  // byte offset is a multiple of 48 -> 16B aligned float4 stores.
  float4* o4 =
      (float4*)(out + ((size_t)slab * NN + (size_t)(n0 + tid * VEC)) * 3);
  o4[0] = make_float4(cosv.x, l2v.x, l1.x, cosv.y);
  o4[1] = make_float4(l2v.y, l1.y, cosv.z, l2v.z);
  o4[2] = make_float4(l1.z, cosv.w, l2v.w, l1.w);
}

extern "C" void kernel_launch(void* const* d_in, const int* in_sizes, int n_in,
                              void* d_out, int out_size, void* d_ws,
                              size_t ws_size, hipStream_t stream) {
  (void)in_sizes; (void)n_in; (void)out_size; (void)d_ws; (void)ws_size;
  const float* x = (const float*)d_in[0];
  const float* y = (const float*)d_in[1];
  float* out = (float*)d_out;
  sim_kernel<<<dim3(SLABS * TILES), dim3(THREADS), 0, stream>>>(x, y, out);
}